// FeatureAttentionLayer_65292092833943
// MI455X (gfx1250) — compile-verified
//
#include <hip/hip_runtime.h>
#include <hip/hip_bf16.h>

typedef __attribute__((ext_vector_type(2))) float v2f;
typedef __attribute__((ext_vector_type(8))) float v8f;

#define ALPHA 0.2f
#define BATCH 32
#define WDIM 128
#define KDIM 1024
#define JT   64            // j-chunk staged in LDS per iteration
#define NCHUNK (KDIM / JT)
#define PSTR 258           // dwords per pair-row (2*128 data + 2 pad): conflict-free stores

// ---------------- Kernel 1: s1/s2 projections + factored exponentials ----------------
__global__ __launch_bounds__(256) void fal_prep_kernel(const float* __restrict__ x,
                                                       const float* __restrict__ w,
                                                       float* __restrict__ s1,
                                                       float* __restrict__ s2,
                                                       float* __restrict__ t1,
                                                       float* __restrict__ ta) {
  int idx = blockIdx.x * 256 + threadIdx.x;       // b*K + k
  int b = idx >> 10;
  int k = idx & (KDIM - 1);
  const float* xp = x + ((size_t)b << 17) + k;    // &x[b,0,k]
  float a1 = 0.f, a2 = 0.f;
#pragma unroll 8
  for (int wi = 0; wi < WDIM; ++wi) {
    float xv = xp[(size_t)wi << 10];              // coalesced across threads (k contiguous)
    a1 = fmaf(xv, w[wi], a1);                     // w[] uniform -> scalar loads
    a2 = fmaf(xv, w[WDIM + wi], a2);
  }
  s1[idx] = a1;
  s2[idx] = a2;
  t1[idx] = __expf(a2);
  ta[idx] = __expf(ALPHA * a2);
}

// ---------------- Kernel 2: per-batch max of s2 ----------------
__global__ __launch_bounds__(256) void fal_rowmax_kernel(const float* __restrict__ s2,
                                                         float* __restrict__ s2max) {
  __shared__ float red[256];
  int b = blockIdx.x;
  float m = -3.4e38f;
  for (int j = threadIdx.x; j < KDIM; j += 256)
    m = fmaxf(m, s2[b * KDIM + j]);
  red[threadIdx.x] = m;
  __syncthreads();
  for (int s = 128; s > 0; s >>= 1) {
    if (threadIdx.x < s) red[threadIdx.x] = fmaxf(red[threadIdx.x], red[threadIdx.x + s]);
    __syncthreads();
  }
  if (threadIdx.x == 0) s2max[b] = red[0];
}

// ---------------- Kernel 3: softmax-weighted aggregation via f32 WMMA ----------------
// grid = (K/128, B), block = 256 (8 waves). Wave w owns output rows [i0, i0+16).
// LDS V tile is k-pair interleaved: element (jj, w) at Vt[(jj>>1)*PSTR + 2*w + (jj&1)],
// so each lane's WMMA B operand is one aligned ds_load_b64.
__global__ __launch_bounds__(256) void fal_attn_kernel(const float* __restrict__ x,
                                                       const float* __restrict__ s1g,
                                                       const float* __restrict__ s2g,
                                                       const float* __restrict__ t1g,
                                                       const float* __restrict__ tag,
                                                       const float* __restrict__ s2mx,
                                                       float* __restrict__ out) {
  __shared__ float  Vt[(JT / 2) * PSTR];   // 32 pair-rows x 258 dwords (~33 KB)
  __shared__ float4 qsm[JT];               // per-j: (s2, exp(s2), exp(alpha*s2), 0)

  const int tid  = threadIdx.x;
  const int lane = tid & 31;
  const int wave = tid >> 5;
  const int half = lane >> 4;              // selects K residues {0,1} vs {2,3}
  const int m    = lane & 15;              // A row (M) / B,C column (N)
  const int b    = blockIdx.y;
  const int i0   = blockIdx.x * 128 + wave * 16;

  // Per-row softmax constants (row = m; identical in both halves)
  const float s1v  = s1g[b * KDIM + i0 + m];
  const float zmax = s1v + s2mx[b];
  const float mrow = fmaxf(zmax, ALPHA * zmax);   // lrelu is monotone -> row max up front
  const float u1   = __expf(s1v - mrow);          // z>0 branch scale
  const float ua   = __expf(ALPHA * s1v - mrow);  // z<=0 branch scale

  v8f acc[8];
#pragma unroll
  for (int t = 0; t < 8; ++t) acc[t] = (v8f)0.f;
  float dsum = 0.f;

  const int k0 = half * 2;

  // ---- staging geometry: thread covers (w0 + 16*it, j0 + 4*jj4 .. +3), it = 0..7 ----
  const int jj4 = tid & 15;                 // j-group of 4
  const int w0  = tid >> 4;                 // base w (0..15)
  const float* gbase = x + (((size_t)b * WDIM + w0) << 10) + (jj4 << 2);
  const int lbase = (jj4 * 2) * PSTR + 2 * w0;   // dword index of (jj=4*jj4, w=w0)

  // preload chunk 0 into registers (software pipeline)
  float4 r[8];
#pragma unroll
  for (int it = 0; it < 8; ++it)
    r[it] = *(const float4*)(gbase + ((size_t)(it * 16) << 10));

  for (int c = 0; c < NCHUNK; ++c) {
    const int j0 = c * JT;
    __syncthreads();                        // previous compute done; LDS free
    // store staged registers: two aligned b64 stores per float4 (conflict-free banks)
#pragma unroll
    for (int it = 0; it < 8; ++it) {
      int o = lbase + it * 32;
      *(float2*)&Vt[o]        = make_float2(r[it].x, r[it].y);   // jj even slot pair
      *(float2*)&Vt[o + PSTR] = make_float2(r[it].z, r[it].w);   // jj+1 pair-row
    }
    if (tid < JT) {
      int j = b * KDIM + j0 + tid;
      qsm[tid] = make_float4(s2g[j], t1g[j], tag[j], 0.f);
    }
    // issue next chunk's global loads now; consumed after the next barrier
    if (c + 1 < NCHUNK) {
#pragma unroll
      for (int it = 0; it < 8; ++it)
        r[it] = *(const float4*)(gbase + ((size_t)(it * 16) << 10) + (c + 1) * JT);
    }
    __syncthreads();

#pragma unroll
    for (int js = 0; js < JT; js += 4) {
      // A = P[16, js:js+4]: lane holds k = k0, k0+1 (16x4 f32 A layout)
      float4 qa = qsm[js + k0];
      float4 qb = qsm[js + k0 + 1];
      float z0 = s1v + qa.x;
      float z1 = s1v + qb.x;
      float p0 = (z0 > 0.f) ? u1 * qa.y : ua * qa.z;   // exp(e-m), factored: no exp here
      float p1 = (z1 > 0.f) ? u1 * qb.y : ua * qb.z;
      dsum += p0 + p1;
      v2f a; a.x = p0; a.y = p1;

      // B rows (js+k0, js+k0+1) are interleaved in pair-row js/2 + half
      const float* vrow = &Vt[(js / 2 + half) * PSTR + 2 * m];
#pragma unroll
      for (int t = 0; t < 8; ++t) {
        v2f bb = *(const v2f*)(vrow + t * 32);          // one ds_load_b64, aligned pair
        acc[t] = __builtin_amdgcn_wmma_f32_16x16x4_f32(false, a, false, bb,
                                                       (short)0, acc[t], false, false);
      }
    }
  }

  // Denominator: lane m covered residues {k0,k0+1}; fold halves (wave32 xor-16)
  float dtot = dsum + __shfl_xor(dsum, 16, 32);

  // Write h = acc / denom. C layout: acc[t][rr] -> row rr + 8*half, col t*16 + m.
#pragma unroll
  for (int rr = 0; rr < 8; ++rr) {
    int row   = rr + 8 * half;
    float d   = __shfl(dtot, rr + 8 * half, 32);
    float inv = 1.0f / d;
    float* orow = out + (((size_t)b * KDIM + i0 + row) << 7) + m;
#pragma unroll
    for (int t = 0; t < 8; ++t)
      orow[t * 16] = acc[t][rr] * inv;
  }
}

// ---------------- Launcher ----------------
extern "C" void kernel_launch(void* const* d_in, const int* in_sizes, int n_in,
                              void* d_out, int out_size, void* d_ws, size_t ws_size,
                              hipStream_t stream) {
  const float* x = (const float*)d_in[0];   // (32,128,1024) f32
  const float* w = (const float*)d_in[1];   // (256,1) f32
  float* out = (float*)d_out;               // (32,1024,128) f32

  float* s1  = (float*)d_ws;                // B*K each
  float* s2  = s1 + BATCH * KDIM;
  float* t1  = s2 + BATCH * KDIM;
  float* ta  = t1 + BATCH * KDIM;
  float* s2m = ta + BATCH * KDIM;           // B

  fal_prep_kernel<<<(BATCH * KDIM) / 256, 256, 0, stream>>>(x, w, s1, s2, t1, ta);
  fal_rowmax_kernel<<<BATCH, 256, 0, stream>>>(s2, s2m);
  dim3 grid(KDIM / 128, BATCH);
  fal_attn_kernel<<<grid, 256, 0, stream>>>(x, s1, s2, t1, ta, s2m, out);
}